// CapsuleLayer_33698313404869
// MI455X (gfx1250) — compile-verified
//
#include <hip/hip_runtime.h>
#include <math.h>

// CapsuleLayer dynamic routing, fused (no u_hat materialization).
// B=64, IU=8, IC=9216, NU=10, US=16, 3 routing iters.
// Heavy contractions: V_WMMA_F32_16X16X4_F32 (fp32, matches reference).
// LDS staging: GLOBAL_LOAD_ASYNC_TO_LDS_B128 + s_wait_asynccnt, double-buffered.

#define ICAP  9216
#define BATCH 64
#define NUNIT 10
#define USZ   16
#define IUNIT 8
#define NROUTE 3

#define NBS 288                  // split-K blocks for s-kernel: 9216/288 = 32 i per block
#define ICHUNK (ICAP / NBS)      // 32
#define GROUP 2                  // i-slices per async stage (double buffered)
#define NGROUP (ICHUNK / GROUP)  // 16
#define NBU 144                  // u_vj blocks: 144*8 waves, 4 i-pairs per wave
#define PAIRS_PER_WAVE 4         // (9216/2) / (144*8) = 4

#define XSLICE (BATCH * IUNIT)         // 512 floats per i
#define WSLICE (NUNIT * USZ * IUNIT)   // 1280 floats per i

typedef float v2f __attribute__((ext_vector_type(2)));
typedef float v8f __attribute__((ext_vector_type(8)));

__device__ inline v8f wmma_f32_16x16x4(v2f a, v2f b, v8f c) {
#if defined(__AMDGCN__)
  // (neg_a, A, neg_b, B, c_mod, C, reuse_a, reuse_b)
  return __builtin_amdgcn_wmma_f32_16x16x4_f32(false, a, false, b, (short)0, c,
                                               false, false);
#else
  (void)a; (void)b;
  return c;
#endif
}

// Async copy 16B global -> LDS, tracked by ASYNCcnt (CDNA5).
// lds_addr: raw LDS byte offset (= low 32 bits of flat shared address).
// saddr: uniform 64-bit base (SGPR pair), voff: per-lane byte offset.
__device__ inline void async_load_b128(unsigned lds_addr, const void* saddr,
                                       unsigned voff) {
#if defined(__AMDGCN__)
  asm volatile("global_load_async_to_lds_b128 %0, %1, %2"
               :: "v"(lds_addr), "v"(voff), "s"(saddr)
               : "memory");
#else
  (void)lds_addr; (void)saddr; (void)voff;
#endif
}

__device__ inline void wait_async0() {
#if defined(__AMDGCN__)
  asm volatile("s_wait_asynccnt 0x0" ::: "memory");
#endif
}

__device__ inline unsigned lds_off(const void* p) {
  return (unsigned)(uintptr_t)p;  // low 32 bits of the flat address are the LDS offset (ISA 10.2)
}

// ---------------------------------------------------------------- utilities
__global__ void zero_kernel(float* __restrict__ p, int n) {
  int t = blockIdx.x * blockDim.x + threadIdx.x;
  if (t < n) p[t] = 0.0f;
}

// x[B][IU][IC] viewed as x2d[r=(b*8+u)][i]  ->  xt[i][b][u] = xt2d[i][r]
__global__ __launch_bounds__(256) void transpose_kernel(
    const float* __restrict__ x, float* __restrict__ xt) {
  __shared__ float tile[32][33];
  const int bx = blockIdx.x;            // i-tile: 0..287
  const int by = blockIdx.y;            // r-tile: 0..15
  const int tx = threadIdx.x & 31;
  const int ty = threadIdx.x >> 5;      // 0..7
#pragma unroll
  for (int k = 0; k < 4; ++k) {
    int r = by * 32 + ty + k * 8;
    tile[ty + k * 8][tx] = x[(size_t)r * ICAP + bx * 32 + tx];
  }
  __syncthreads();
#pragma unroll
  for (int k = 0; k < 4; ++k) {
    int i = bx * 32 + ty + k * 8;
    int r = by * 32 + tx;
    xt[(size_t)i * XSLICE + r] = tile[tx][ty + k * 8];
  }
}

// ------------------------------------------------- softmax over IC (axis 0)
__global__ __launch_bounds__(256) void colsoftmax_reduce_kernel(
    const float* __restrict__ b_ij, float* __restrict__ maxden) {
  __shared__ float red[256];
  const int j = blockIdx.x;
  float m = -3.4e38f;
  for (int i = threadIdx.x; i < ICAP; i += 256)
    m = fmaxf(m, b_ij[i * NUNIT + j]);
  red[threadIdx.x] = m;
  __syncthreads();
  for (int st = 128; st > 0; st >>= 1) {
    if (threadIdx.x < st)
      red[threadIdx.x] = fmaxf(red[threadIdx.x], red[threadIdx.x + st]);
    __syncthreads();
  }
  const float mx = red[0];
  __syncthreads();
  float sum = 0.0f;
  for (int i = threadIdx.x; i < ICAP; i += 256)
    sum += expf(b_ij[i * NUNIT + j] - mx);
  red[threadIdx.x] = sum;
  __syncthreads();
  for (int st = 128; st > 0; st >>= 1) {
    if (threadIdx.x < st) red[threadIdx.x] += red[threadIdx.x + st];
    __syncthreads();
  }
  if (threadIdx.x == 0) {
    maxden[j] = mx;
    maxden[NUNIT + j] = red[0];
  }
}

__global__ void c_kernel(const float* __restrict__ b_ij,
                         const float* __restrict__ maxden,
                         float* __restrict__ c) {
  int t = blockIdx.x * blockDim.x + threadIdx.x;
  if (t < ICAP * NUNIT) {
    int j = t % NUNIT;
    c[t] = expf(b_ij[t] - maxden[j]) / maxden[NUNIT + j];
  }
}

// ---------------------------------------------------------------- s-kernel
// s[b,(j,s)] = sum_i sum_u xt[i][b][u] * (c[i,j]*W[i,j,s,u])
// WMMA: M=batch(4 tiles of 16), N=(j,s)(tile==j), K=u(2 steps of 4).
// Split-K over IC blocks; async double-buffered LDS staging of x/W slices.
__global__ __launch_bounds__(256) void sj_kernel(
    const float* __restrict__ xt, const float* __restrict__ w,
    const float* __restrict__ c, float* __restrict__ part) {
  __shared__ __align__(16) float xsb[2][GROUP * XSLICE];   // 2*1024 floats
  __shared__ __align__(16) float wlb[2][GROUP * WSLICE];   // 2*2560 floats

  const int tid = threadIdx.x;
  const int lane = tid & 31;
  const int wv = tid >> 5;              // 0..7
  const int Lm = lane & 15;
  const int half = lane >> 4;
  const int kb = 2 * half;              // K sub-offset per A/B layout
  const int mtile = wv >> 1;            // 0..3
  const int jbase = (wv & 1) * 5;       // 0 or 5
  const int ibase = blockIdx.x * ICHUNK;

  // Issue one group's async loads into buffer `buf`.
  auto issue = [&](int g, int buf) {
    const size_t i0 = (size_t)(ibase + g * GROUP);
    // x: GROUP*512 floats = 256 b128 per block (1 per thread)
    for (int t = tid; t < GROUP * XSLICE / 4; t += 256)
      async_load_b128(lds_off(&xsb[buf][t * 4]), xt,
                      (unsigned)((i0 * XSLICE + t * 4) * sizeof(float)));
    // W: GROUP*1280 floats = 640 b128 per block
    for (int t = tid; t < GROUP * WSLICE / 4; t += 256)
      async_load_b128(lds_off(&wlb[buf][t * 4]), w,
                      (unsigned)((i0 * WSLICE + t * 4) * sizeof(float)));
  };

  v8f acc[5];
#pragma unroll
  for (int t = 0; t < 5; ++t) acc[t] = {};

  issue(0, 0);  // prologue
  for (int g = 0; g < NGROUP; ++g) {
    const int cur = g & 1;
    wait_async0();       // my async writes for group g are in LDS
    __syncthreads();     // everyone's writes visible; buf[cur^1] free to refill
    if (g + 1 < NGROUP) issue(g + 1, cur ^ 1);

    for (int sl = 0; sl < GROUP; ++sl) {
      const int i = ibase + g * GROUP + sl;
      const float* xs = &xsb[cur][sl * XSLICE];
      const float* wl = &wlb[cur][sl * WSLICE];
      float cj[5];
#pragma unroll
      for (int t = 0; t < 5; ++t) cj[t] = c[i * NUNIT + jbase + t];
#pragma unroll
      for (int ks = 0; ks < 2; ++ks) {
        const int k0 = ks * 4 + kb;     // u index for this lane-half
        v2f a;
        a.x = xs[(mtile * 16 + Lm) * IUNIT + k0];
        a.y = xs[(mtile * 16 + Lm) * IUNIT + k0 + 1];
#pragma unroll
        for (int t = 0; t < 5; ++t) {
          const int j = jbase + t;
          v2f bb;
          bb.x = cj[t] * wl[(j * USZ + Lm) * IUNIT + k0];
          bb.y = cj[t] * wl[(j * USZ + Lm) * IUNIT + k0 + 1];
          acc[t] = wmma_f32_16x16x4(a, bb, acc[t]);
        }
      }
    }
    __syncthreads();     // done reading buf[cur] before next refill targets it
  }

  float* pb = part + (size_t)blockIdx.x * (BATCH * NUNIT * USZ);
#pragma unroll
  for (int t = 0; t < 5; ++t) {
    const int j = jbase + t;
#pragma unroll
    for (int r = 0; r < 8; ++r) {
      const int M = mtile * 16 + r + 8 * half;  // batch index
      pb[M * (NUNIT * USZ) + j * USZ + Lm] = acc[t][r];
    }
  }
}

__global__ void sreduce_kernel(const float* __restrict__ part,
                               float* __restrict__ s) {
  int t = blockIdx.x * blockDim.x + threadIdx.x;
  if (t < BATCH * NUNIT * USZ) {
    float acc = 0.0f;
    for (int k = 0; k < NBS; ++k)       // fixed order -> deterministic
      acc += part[(size_t)k * (BATCH * NUNIT * USZ) + t];
    s[t] = acc;
  }
}

// squash: norm over NU axis per (b, s16)
__global__ void squash_kernel(const float* __restrict__ s,
                              float* __restrict__ v, float* __restrict__ out) {
  int t = blockIdx.x * blockDim.x + threadIdx.x;
  if (t < BATCH * USZ) {
    const int b = t / USZ, s16 = t % USZ;
    const float* sp = s + b * NUNIT * USZ + s16;
    float msq = 0.0f;
#pragma unroll
    for (int j = 0; j < NUNIT; ++j) {
      float x = sp[j * USZ];
      msq += x * x;
    }
    const float scale = (msq / (1.0f + msq)) / sqrtf(msq);
#pragma unroll
    for (int j = 0; j < NUNIT; ++j) {
      const float val = sp[j * USZ] * scale;
      v[b * NUNIT * USZ + j * USZ + s16] = val;
      out[b * NUNIT * USZ + j * USZ + s16] = val;
    }
  }
}

// ---------------------------------------------------------------- u_vj
// Per i-pair (i0,i1): xv[(ii,u)][(j,s)] = sum_b xt[i][b][u]*v[b][(j,s)]
// WMMA: M=16 (2 i's x 8 u), N=(j,s) 10 tiles, K=b (16 steps of 4).
// Then u_vj[i,j] = (1/64) * sum_{u,s} W[i,j,s,u]*xv -> b_ij[i,j] += u_vj.
__global__ __launch_bounds__(256) void uvj_kernel(
    const float* __restrict__ xt, const float* __restrict__ w,
    const float* __restrict__ v, float* __restrict__ b_ij) {
  __shared__ __align__(16) float vl[BATCH * NUNIT * USZ];   // 40KB
  const int tid = threadIdx.x;
  // async stage v (2560 b128 = 10 per thread, uniform)
  for (int t = tid; t < (BATCH * NUNIT * USZ) / 4; t += 256)
    async_load_b128(lds_off(&vl[t * 4]), v, (unsigned)(t * 16));
  wait_async0();
  __syncthreads();

  const int lane = tid & 31;
  const int wv = tid >> 5;
  const int Lm = lane & 15;
  const int half = lane >> 4;
  const int kb = 2 * half;
  const int gw = blockIdx.x * 8 + wv;   // global wave id: 0..1151
  const int iiA = Lm >> 3;              // A-frag: which i of the pair
  const int uA = Lm & 7;                // A-frag: u index

  for (int pp = 0; pp < PAIRS_PER_WAVE; ++pp) {
    const int p = gw * PAIRS_PER_WAVE + pp;
    const int i0 = 2 * p, i1 = 2 * p + 1;
    const int iA = iiA ? i1 : i0;
    v8f acc[NUNIT];
#pragma unroll
    for (int j = 0; j < NUNIT; ++j) acc[j] = {};
#pragma unroll 4
    for (int ks = 0; ks < 16; ++ks) {
      const int k0 = ks * 4 + kb;       // batch index for this lane-half
      v2f a;
      a.x = xt[((size_t)iA * BATCH + k0) * IUNIT + uA];
      a.y = xt[((size_t)iA * BATCH + k0 + 1) * IUNIT + uA];
#pragma unroll
      for (int j = 0; j < NUNIT; ++j) {
        v2f bb;
        bb.x = vl[k0 * (NUNIT * USZ) + j * USZ + Lm];
        bb.y = vl[(k0 + 1) * (NUNIT * USZ) + j * USZ + Lm];
        acc[j] = wmma_f32_16x16x4(a, bb, acc[j]);
      }
    }
    // D layout: reg r, lanes<16 -> (i0, u=r); lanes>=16 -> (i1, u=r); N=Lm=s
    const int iD = half ? i1 : i0;
#pragma unroll
    for (int j = 0; j < NUNIT; ++j) {
      const float* wp = w + ((size_t)(iD * NUNIT + j) * USZ + Lm) * IUNIT;
      float t = 0.0f;
#pragma unroll
      for (int r = 0; r < 8; ++r) t += wp[r] * acc[j][r];
      t += __shfl_xor(t, 1, 16);
      t += __shfl_xor(t, 2, 16);
      t += __shfl_xor(t, 4, 16);
      t += __shfl_xor(t, 8, 16);
      if (Lm == 0)  // unique (i,j) owner -> race-free RMW
        b_ij[iD * NUNIT + j] += t * (1.0f / (float)BATCH);
    }
  }
}

// ---------------------------------------------------------------- launcher
extern "C" void kernel_launch(void* const* d_in, const int* in_sizes, int n_in,
                              void* d_out, int out_size, void* d_ws,
                              size_t ws_size, hipStream_t stream) {
  (void)in_sizes; (void)n_in; (void)out_size; (void)ws_size;
  const float* x = (const float*)d_in[0];   // [64, 8, 9216]
  const float* w = (const float*)d_in[1];   // [1, 9216, 10, 16, 8]
  float* out = (float*)d_out;               // [64, 10, 16, 1]

  float* p = (float*)d_ws;                  // ~31.5 MB total
  float* xt   = p; p += (size_t)ICAP * XSLICE;             // 4,718,592
  float* b_ij = p; p += ICAP * NUNIT;                      // 92,160
  float* mxdn = p; p += 32;                                // max[10], denom[10]
  float* cbuf = p; p += ICAP * NUNIT;                      // 92,160
  float* sbuf = p; p += BATCH * NUNIT * USZ;               // 10,240
  float* vbuf = p; p += BATCH * NUNIT * USZ;               // 10,240
  float* part = p;                                         // 288*10,240

  zero_kernel<<<(ICAP * NUNIT + 255) / 256, 256, 0, stream>>>(b_ij,
                                                              ICAP * NUNIT);
  transpose_kernel<<<dim3(ICAP / 32, (BATCH * IUNIT) / 32), 256, 0, stream>>>(
      x, xt);

  for (int it = 0; it < NROUTE; ++it) {
    colsoftmax_reduce_kernel<<<NUNIT, 256, 0, stream>>>(b_ij, mxdn);
    c_kernel<<<(ICAP * NUNIT + 255) / 256, 256, 0, stream>>>(b_ij, mxdn, cbuf);
    sj_kernel<<<NBS, 256, 0, stream>>>(xt, w, cbuf, part);
    sreduce_kernel<<<(BATCH * NUNIT * USZ + 255) / 256, 256, 0, stream>>>(part,
                                                                          sbuf);
    squash_kernel<<<(BATCH * USZ + 255) / 256, 256, 0, stream>>>(sbuf, vbuf,
                                                                 out);
    if (it < NROUTE - 1)
      uvj_kernel<<<NBU, 256, 0, stream>>>(xt, w, vbuf, b_ij);
  }
}